// VectorQuantizer_17188459119253
// MI455X (gfx1250) — compile-verified
//
#include <hip/hip_runtime.h>
#include <hip/hip_bf16.h>

// ---------------------------------------------------------------------------
// VQ-VAE vector quantizer for MI455X (gfx1250).
// B=32, C=64, L=4096, K=512.  N = B*L = 131072 rows.
// Core GEMM (N x K, k-dim C=64) done with V_WMMA_F32_16X16X4_F32 (exact f32,
// needed because codebook entries are ~1/512 and argmin gaps are tiny).
// ---------------------------------------------------------------------------

typedef __attribute__((ext_vector_type(2))) float v2f;
typedef __attribute__((ext_vector_type(4))) float v4f;
typedef __attribute__((ext_vector_type(8))) float v8f;

#define VQ_B 32
#define VQ_C 64
#define VQ_L 4096
#define VQ_K 512
#define VQ_N (VQ_B * VQ_L)          // 131072 rows
#define VQ_QELEMS (VQ_B * VQ_C * VQ_L) // 8388608

// ---------------------------------------------------------------------------
// Kernel 0: zero histogram counts, copy embed to output slot.
// ---------------------------------------------------------------------------
__global__ __launch_bounds__(256) void vq_init_copy(const float* __restrict__ embed,
                                                    float* __restrict__ out_embed,
                                                    int* __restrict__ counts) {
    int i = blockIdx.x * 256 + threadIdx.x;
    if (i < VQ_K) counts[i] = 0;
    if (i < VQ_K * VQ_C) out_embed[i] = embed[i];
}

// ---------------------------------------------------------------------------
// Kernel 1: WMMA f32 GEMM + fused argmin.
// Block = 256 threads (8 waves). Each wave computes 16 rows x 512 codes.
// Codebook (512x64 f32 = 128KB) + 0.5*||e||^2 staged in LDS.
//
// A layout (16x4 f32, ISA 7.12.2): lane = M row (mod 16); VGPR0 = K=0 (lanes
// 0-15) / K=2 (lanes 16-31); VGPR1 = K=1 / K=3.
// B layout (4x16 f32, symmetric): lane = N col (mod 16); VGPR0 = K=0 / K=2;
// VGPR1 = K=1 / K=3.  -> lane n reads embed[n][c0 + 2*half + {0,1}].
// C/D layout: VGPR j on lanes 0-15 = (M=j, N=lane); lanes 16-31 = (M=j+8).
// ---------------------------------------------------------------------------
__global__ __launch_bounds__(256) void vq_argmin(const float* __restrict__ x,
                                                 const float* __restrict__ embed,
                                                 int* __restrict__ idx_out) {
    extern __shared__ float smem[];
    float* lds_embed = smem;              // VQ_K * VQ_C floats
    float* lds_esqh  = smem + VQ_K * VQ_C; // VQ_K floats

    const int t = threadIdx.x;

    // Cooperative load of codebook into LDS (float4, coalesced).
    for (int i = t * 4; i < VQ_K * VQ_C; i += 256 * 4) {
        v4f v = *(const v4f*)(embed + i);
        *(v4f*)(lds_embed + i) = v;
    }
    __syncthreads();

    // 0.5 * ||e||^2 per code.
    for (int k = t; k < VQ_K; k += 256) {
        float s = 0.f;
        #pragma unroll 8
        for (int c = 0; c < VQ_C; ++c) {
            float e = lds_embed[k * VQ_C + c];
            s += e * e;
        }
        lds_esqh[k] = 0.5f * s;
    }
    __syncthreads();

    const int wave = t >> 5;        // 0..7
    const int lane = t & 31;
    const int half = lane >> 4;     // 0 | 1
    const int l16  = lane & 15;

    const int row_base = blockIdx.x * 128 + wave * 16;   // 16 rows / wave
    const int n_row = row_base + l16;                    // this lane's A row
    const int b = n_row / VQ_L;                          // rows in a tile share b
    const int l = n_row % VQ_L;
    const float* xrow = x + (size_t)b * VQ_C * VQ_L + l; // element c at xrow[c*L]

    // Load A tile: a[kk] holds k-dim elements c = 4*kk + 2*half + {0,1}.
    v2f a[16];
    #pragma unroll
    for (int kk = 0; kk < 16; ++kk) {
        int c0 = 4 * kk + 2 * half;
        v2f av;
        av.x = xrow[(size_t)c0 * VQ_L];
        av.y = xrow[(size_t)(c0 + 1) * VQ_L];
        a[kk] = av;
    }

    float best[8];
    int   bidx[8];
    #pragma unroll
    for (int j = 0; j < 8; ++j) { best[j] = -3.0e38f; bidx[j] = 0; }

    // 32 column tiles of 16 codes each.
    for (int ct = 0; ct < VQ_K / 16; ++ct) {
        const int ncode = ct * 16 + l16;
        const float* bcol = lds_embed + ncode * VQ_C + 2 * half;

        v8f acc = {0.f, 0.f, 0.f, 0.f, 0.f, 0.f, 0.f, 0.f};
        #pragma unroll
        for (int kk = 0; kk < 16; ++kk) {
            v2f bv;
            bv.x = bcol[4 * kk + 0];
            bv.y = bcol[4 * kk + 1];
            // D = A*B + C  (exact f32 matrix pipe)
            acc = __builtin_amdgcn_wmma_f32_16x16x4_f32(
                false, a[kk], false, bv, (short)0, acc, false, false);
        }

        const float esqh = lds_esqh[ncode];
        #pragma unroll
        for (int j = 0; j < 8; ++j) {
            float s = acc[j] - esqh;   // maximize x.e - 0.5||e||^2  <=> argmin dist
            bool take = (s > best[j]) | ((s == best[j]) & (ncode < bidx[j]));
            best[j] = take ? s : best[j];
            bidx[j] = take ? ncode : bidx[j];
        }
    }

    // Reduce across the 16 lanes of each half (each lane holds one N-residue).
    #pragma unroll
    for (int off = 1; off < 16; off <<= 1) {
        #pragma unroll
        for (int j = 0; j < 8; ++j) {
            float s2 = __shfl_xor(best[j], off, 32);
            int   i2 = __shfl_xor(bidx[j], off, 32);
            bool take = (s2 > best[j]) | ((s2 == best[j]) & (i2 < bidx[j]));
            best[j] = take ? s2 : best[j];
            bidx[j] = take ? i2 : bidx[j];
        }
    }

    if (l16 == 0) {
        #pragma unroll
        for (int j = 0; j < 8; ++j) {
            int r = row_base + j + 8 * half;   // half=0 -> M=j, half=1 -> M=j+8
            idx_out[r] = bidx[j];
        }
    }
}

// ---------------------------------------------------------------------------
// Kernel 2: quantized output (B,C,L), indices-as-float, histogram, and
// per-block partial sum of (q - x)^2 (deterministic tree reduction).
// 512 blocks x 256 threads, one thread per row.
// ---------------------------------------------------------------------------
__global__ __launch_bounds__(256) void vq_output(const float* __restrict__ x,
                                                 const float* __restrict__ embed,
                                                 const int* __restrict__ idx,
                                                 float* __restrict__ quant_out,
                                                 float* __restrict__ idxf_out,
                                                 int* __restrict__ counts,
                                                 float* __restrict__ partials) {
    const int n = blockIdx.x * 256 + threadIdx.x;   // row (b*L + l)
    const int b = n / VQ_L;
    const int l = n % VQ_L;
    const int k = idx[n];

    idxf_out[n] = (float)k;
    atomicAdd(&counts[k], 1);   // integer atomics: order-independent

    const float* e = embed + k * VQ_C;
    const size_t base = (size_t)b * VQ_C * VQ_L + l;
    float sumsq = 0.f;
    #pragma unroll 8
    for (int c = 0; c < VQ_C; ++c) {
        float q  = e[c];
        float xv = x[base + (size_t)c * VQ_L];   // coalesced over l
        float d  = q - xv;
        sumsq += d * d;
        quant_out[base + (size_t)c * VQ_L] = q;  // straight-through == quantized
    }

    __shared__ float red[256];
    red[threadIdx.x] = sumsq;
    __syncthreads();
    for (int s = 128; s > 0; s >>= 1) {
        if (threadIdx.x < s) red[threadIdx.x] += red[threadIdx.x + s];
        __syncthreads();
    }
    if (threadIdx.x == 0) partials[blockIdx.x] = red[0];
}

// ---------------------------------------------------------------------------
// Kernel 3: one-hot encodings (N x K = 268 MB) with non-temporal stores
// (larger than the 192 MB L2 -> bypass).
// ---------------------------------------------------------------------------
__global__ __launch_bounds__(256) void vq_onehot(const int* __restrict__ idx,
                                                 float* __restrict__ enc) {
    const long long i = (long long)blockIdx.x * 256 + threadIdx.x; // float4 index
    const long long total4 = (long long)VQ_N * VQ_K / 4;
    if (i >= total4) return;
    const long long e0 = i * 4;
    const int n  = (int)(e0 >> 9);    // /512
    const int k0 = (int)(e0 & 511);
    const int kk = idx[n];
    v4f v;
    v.x = (k0 + 0 == kk) ? 1.f : 0.f;
    v.y = (k0 + 1 == kk) ? 1.f : 0.f;
    v.z = (k0 + 2 == kk) ? 1.f : 0.f;
    v.w = (k0 + 3 == kk) ? 1.f : 0.f;
    __builtin_nontemporal_store(v, (v4f*)(enc + e0));
}

// ---------------------------------------------------------------------------
// Kernel 4: deterministic finalize — loss and perplexity.
// ---------------------------------------------------------------------------
__global__ __launch_bounds__(256) void vq_finalize(const int* __restrict__ counts,
                                                   const float* __restrict__ partials,
                                                   float* __restrict__ out_loss,
                                                   float* __restrict__ out_perp) {
    __shared__ float red[256];
    const int t = threadIdx.x;

    // loss = (1 + 0.25) * mean((q - x)^2)
    float s = partials[t] + partials[t + 256];   // 512 partials
    red[t] = s;
    __syncthreads();
    for (int st = 128; st > 0; st >>= 1) {
        if (t < st) red[t] += red[t + st];
        __syncthreads();
    }
    const float lossv = 1.25f * red[0] / (float)VQ_QELEMS;
    __syncthreads();

    // perplexity = exp(-sum p*log(p + 1e-10)), p = counts / N
    const float invN = 1.0f / (float)VQ_N;
    float h = 0.f;
    for (int k = t; k < VQ_K; k += 256) {
        float p = (float)counts[k] * invN;
        h += p * logf(p + 1e-10f);
    }
    red[t] = h;
    __syncthreads();
    for (int st = 128; st > 0; st >>= 1) {
        if (t < st) red[t] += red[t + st];
        __syncthreads();
    }
    if (t == 0) {
        *out_loss = lossv;
        *out_perp = expf(-red[0]);
    }
}

// ---------------------------------------------------------------------------
// Launcher.  Output tuple, concatenated flat:
//   [0]                loss                              (1)
//   [1 .. ]            quantized_st (B,C,L)              (8388608)
//   [+1]               perplexity                        (1)
//   [+32768]           embed (K,C)                       (32768)
//   [+131072]          encoding_indices (B,L) as float   (131072)
//   [+67108864]        encodings (N,K) one-hot           (67108864)
// ---------------------------------------------------------------------------
extern "C" void kernel_launch(void* const* d_in, const int* in_sizes, int n_in,
                              void* d_out, int out_size, void* d_ws, size_t ws_size,
                              hipStream_t stream) {
    const float* x     = (const float*)d_in[0];   // (B, C, L) f32
    const float* embed = (const float*)d_in[1];   // (K, C)    f32

    float* out = (float*)d_out;
    float* out_loss  = out;
    float* out_quant = out + 1;
    float* out_perp  = out + 1 + (size_t)VQ_QELEMS;
    float* out_embed = out_perp + 1;
    float* out_idx   = out_embed + (size_t)VQ_K * VQ_C;
    float* out_enc   = out_idx + (size_t)VQ_N;

    int*   ws_idx    = (int*)d_ws;                 // N ints
    int*   ws_counts = ws_idx + VQ_N;              // K ints
    float* ws_part   = (float*)(ws_counts + VQ_K); // 512 floats

    // 0) zero counts + copy embed
    vq_init_copy<<<(VQ_K * VQ_C + 255) / 256, 256, 0, stream>>>(embed, out_embed, ws_counts);

    // 1) WMMA GEMM + argmin (dynamic LDS: 512*64 + 512 floats = 133120 B)
    const size_t lds_bytes = (size_t)(VQ_K * VQ_C + VQ_K) * sizeof(float);
    vq_argmin<<<VQ_N / 128, 256, lds_bytes, stream>>>(x, embed, ws_idx);

    // 2) quantized + indices + histogram + loss partials
    vq_output<<<VQ_N / 256, 256, 0, stream>>>(x, embed, ws_idx, out_quant, out_idx,
                                              ws_counts, ws_part);

    // 3) one-hot encodings (non-temporal)
    const long long total4 = (long long)VQ_N * VQ_K / 4;
    vq_onehot<<<(unsigned)((total4 + 255) / 256), 256, 0, stream>>>(ws_idx, out_enc);

    // 4) finalize scalars
    vq_finalize<<<1, 256, 0, stream>>>(ws_counts, ws_part, out_loss, out_perp);
}